// CenterTOpEXnew_5617817223885
// MI455X (gfx1250) — compile-verified
//
#include <hip/hip_runtime.h>

#define B_SZ 8
#define C_SZ 128
#define NPTS 65536

typedef __attribute__((ext_vector_type(2))) float v2f;
typedef __attribute__((ext_vector_type(8))) float v8f;

__device__ __forceinline__ float wave_sum(float v) {
    #pragma unroll
    for (int o = 16; o > 0; o >>= 1) v += __shfl_xor(v, o, 32);
    return v;
}

// ---------------------------------------------------------------------------
// Per-iteration: centers <- EMA update (or init), Cn <- row-normalized centers
// grid(B), block(128)
__global__ void prep_centers(const float* __restrict__ cinit, float* centers,
                             float* cn, const float* __restrict__ citer, int first) {
    int b = blockIdx.x, t = threadIdx.x;
    __shared__ float red[128];
    for (int r = 0; r < 2; ++r) {
        float* crow = centers + (b * 2 + r) * 128;
        float v;
        if (first) {
            v = cinit[r * 128 + t];
        } else {
            float c0 = crow[t];
            v = c0 + 0.1f * (citer[(b * 2 + r) * 128 + t] - c0);
        }
        crow[t] = v;
        red[t] = v * v;
        __syncthreads();
        for (int s = 64; s > 0; s >>= 1) {
            if (t < s) red[t] += red[t + s];
            __syncthreads();
        }
        float rn = fmaxf(sqrtf(red[0]), 1e-12f);
        cn[(b * 2 + r) * 128 + t] = v / rn;
        __syncthreads();
    }
}

// ---------------------------------------------------------------------------
// Dist pass: WMMA f32 16x16x4. grid(512, B), block(256) = 8 waves x 16 rows.
// Writes dists [B][N][2], labels [B][N], optional labelPinit, block partials
// (sum labels, sum dTh0, sum dTh1) per (batch, block).
__global__ void dist_kernel(const float* __restrict__ F, const float* __restrict__ cn,
                            float* __restrict__ dists, float* __restrict__ labels,
                            float* labelP, float* p1) {
    int b = blockIdx.y;
    int tid = threadIdx.x, lane = tid & 31, w = tid >> 5;
    __shared__ float cnl[256];
    __shared__ float dl[8][32];
    __shared__ float pr[8][3];
    cnl[tid] = cn[b * 256 + tid];
    __syncthreads();

    const float* Fb = F + (size_t)b * C_SZ * NPTS;
    int n = lane & 15, h = lane >> 4;
    int row0 = (blockIdx.x * 8 + w) * 16;
    const float* pa = Fb + (row0 + n);

    v8f acc = {0.f, 0.f, 0.f, 0.f, 0.f, 0.f, 0.f, 0.f};
    float ssp = 0.f;
    int nn = (n < 2) ? n : 0;   // safe LDS index for B-operand
    #pragma unroll 4
    for (int k = 0; k < 32; ++k) {
        int c0 = 4 * k + 2 * h;
        float a0 = pa[(size_t)c0 * NPTS];
        float a1 = pa[(size_t)(c0 + 1) * NPTS];
        ssp += a0 * a0 + a1 * a1;
        float bv0 = cnl[nn * 128 + 4 * k + h];
        float bv1 = cnl[nn * 128 + 4 * k + 2 + h];
        v2f A = {a0, a1};
        v2f Bv = {(n < 2) ? bv0 : 0.f, (n < 2) ? bv1 : 0.f};
        acc = __builtin_amdgcn_wmma_f32_16x16x4_f32(false, A, false, Bv,
                                                    (short)0, acc, false, false);
    }
    float ss = ssp + __shfl_xor(ssp, 16, 32);

    // stage C columns 0,1 (dot0, dot1 per row) to LDS for thread-per-row finish
    if (n < 2) {
        #pragma unroll
        for (int r = 0; r < 8; ++r) dl[w][(r + 8 * h) * 2 + n] = acc[r];
    }

    float d0 = 0.f, d1 = 0.f, lf = 0.f;
    if (lane < 16) {
        float inv = 1.0f / fmaxf(sqrtf(ss), 1e-12f);
        float dot0 = dl[w][lane * 2 + 0];
        float dot1 = dl[w][lane * 2 + 1];
        d0 = 0.5f * (1.0f - dot0 * inv);
        d1 = 0.5f * (1.0f - dot1 * inv);
        lf = (d1 < d0) ? 1.0f : 0.0f;   // argmin, ties -> 0
        size_t gi = (size_t)b * NPTS + row0 + lane;
        float2 dv; dv.x = d0; dv.y = d1;
        *reinterpret_cast<float2*>(dists + gi * 2) = dv;
        labels[gi] = lf;
        if (labelP) labelP[gi] = lf;
    }

    float s0 = wave_sum(lf);
    float s1 = wave_sum((lf == 0.f) ? d0 : 0.f);   // dTh[:,0] (d0,d1 are 0 on lanes>=16)
    float s2 = wave_sum(d1 * lf);                  // dTh[:,1]
    if (lane == 0) { pr[w][0] = s0; pr[w][1] = s1; pr[w][2] = s2; }
    __syncthreads();
    if (tid == 0) {
        float a = 0.f, bb = 0.f, c = 0.f;
        for (int ww = 0; ww < 8; ++ww) { a += pr[ww][0]; bb += pr[ww][1]; c += pr[ww][2]; }
        float* o = p1 + ((size_t)b * 512 + blockIdx.x) * 3;
        o[0] = a; o[1] = bb; o[2] = c;
    }
}

// ---------------------------------------------------------------------------
// Reduce dist-pass partials -> per-batch thresholds. grid(B), block(256)
__global__ void thresh_kernel(const float* __restrict__ p1, float* thr,
                              const int* pn1, const int* pn2) {
    int b = blockIdx.x, t = threadIdx.x;
    __shared__ float r0[256], r1[256], r2[256];
    float a = 0.f, bb = 0.f, c = 0.f;
    for (int k = t; k < 512; k += 256) {
        const float* p = p1 + ((size_t)b * 512 + k) * 3;
        a += p[0]; bb += p[1]; c += p[2];
    }
    r0[t] = a; r1[t] = bb; r2[t] = c;
    __syncthreads();
    for (int s = 128; s > 0; s >>= 1) {
        if (t < s) { r0[t] += r0[t + s]; r1[t] += r1[t + s]; r2[t] += r2[t + s]; }
        __syncthreads();
    }
    if (t == 0) {
        float sl = r0[0], sd0 = r1[0], sd1 = r2[0];
        float chgNum = sl + 1.0f;
        float unchgNum = ((float)NPTS - sl) + 1.0f;
        thr[b * 2 + 0] = (sd0 / unchgNum) * (float)pn2[0];   // unchg threshold
        thr[b * 2 + 1] = (sd1 / chgNum) / (float)pn1[0];     // chg threshold
    }
}

// ---------------------------------------------------------------------------
// Flag-weighted column sums. grid(32, B), block(256); each block covers 2048
// rows (8 register-cached chunks). Deterministic butterfly + ordered combine.
__global__ void flagsum_kernel(const float* __restrict__ F, const float* __restrict__ dists,
                               const float* __restrict__ thr, float* p3) {
    int b = blockIdx.y, tid = threadIdx.x, lane = tid & 31, w = tid >> 5;
    const float* Fb = F + (size_t)b * C_SZ * NPTS;
    float thrU = thr[b * 2 + 0], thrC = thr[b * 2 + 1];
    int base = blockIdx.x * 2048;

    float fu[8], fc[8];
    int idx[8];
    float sfu = 0.f, sfc = 0.f;
    #pragma unroll
    for (int k = 0; k < 8; ++k) {
        int i = base + k * 256 + tid;
        idx[k] = i;
        float2 dv = *reinterpret_cast<const float2*>(dists + ((size_t)b * NPTS + i) * 2);
        float lf = (dv.y < dv.x) ? 1.f : 0.f;
        float c_ = (lf * dv.y > thrC) ? 0.f : lf;
        float u_ = ((1.f - lf) * dv.x > thrU) ? 0.f : (1.f - lf);
        fc[k] = c_; fu[k] = u_; sfc += c_; sfu += u_;
    }

    float au[4] = {0.f, 0.f, 0.f, 0.f}, ac[4] = {0.f, 0.f, 0.f, 0.f};
    for (int c = 0; c < 128; ++c) {
        const float* col = Fb + (size_t)c * NPTS;
        float su = 0.f, sc = 0.f;
        #pragma unroll
        for (int k = 0; k < 8; ++k) {
            float x = col[idx[k]];
            su += x * fu[k];
            sc += x * fc[k];
        }
        su = wave_sum(su);
        sc = wave_sum(sc);
        int j = c >> 5;
        if ((c & 31) == lane) { au[j] += su; ac[j] += sc; }
    }

    __shared__ float lds[8][256];
    __shared__ float lfs[8][2];
    #pragma unroll
    for (int j = 0; j < 4; ++j) {
        lds[w][j * 32 + lane] = au[j];
        lds[w][128 + j * 32 + lane] = ac[j];
    }
    sfu = wave_sum(sfu);
    sfc = wave_sum(sfc);
    if (lane == 0) { lfs[w][0] = sfu; lfs[w][1] = sfc; }
    __syncthreads();
    float s = 0.f;
    for (int ww = 0; ww < 8; ++ww) s += lds[ww][tid];
    float* o = p3 + ((size_t)b * 32 + blockIdx.x) * 258;
    o[tid] = s;
    if (tid < 2) {
        float fs = 0.f;
        for (int ww = 0; ww < 8; ++ww) fs += lfs[ww][tid];
        o[256 + tid] = fs;
    }
}

// ---------------------------------------------------------------------------
// Reduce flag partials -> centersIter[b]; at first iter also cosine-sim vs init.
// grid(B), block(256)
__global__ void citer_kernel(const float* __restrict__ p3, float* citer,
                             const float* __restrict__ cinit, float* cini, int first) {
    int b = blockIdx.x, t = threadIdx.x;
    __shared__ float sums[258];
    float s = 0.f;
    for (int k = 0; k < 32; ++k) s += p3[((size_t)b * 32 + k) * 258 + t];
    sums[t] = s;
    if (t < 2) {
        float fs = 0.f;
        for (int k = 0; k < 32; ++k) fs += p3[((size_t)b * 32 + k) * 258 + 256 + t];
        sums[256 + t] = fs;
    }
    __syncthreads();
    float denom = (t < 128) ? (sums[256] + 1.f) : (sums[257] + 1.f);
    float ci = sums[t] / denom;                 // t<128: unchg row; else chg row
    citer[(size_t)b * 256 + t] = ci;
    if (first) {
        __syncthreads();
        sums[t] = ci;
        __syncthreads();
        if (t < 2) {
            const float* a = &sums[t * 128];
            const float* c0 = cinit + t * 128;
            float num = 0.f, na = 0.f, nb = 0.f;
            for (int k = 0; k < 128; ++k) {
                num += a[k] * c0[k]; na += a[k] * a[k]; nb += c0[k] * c0[k];
            }
            float den = fmaxf(sqrtf(na) * sqrtf(nb), 1e-8f);
            cini[b * 2 + t] = num / den;
        }
    }
}

// ---------------------------------------------------------------------------
__global__ void finish_kernel(const float* __restrict__ citer, const float* __restrict__ cini,
                              float* outC, float* outCini) {
    int t = threadIdx.x;
    float s = 0.f;
    for (int b = 0; b < B_SZ; ++b) s += citer[(size_t)b * 256 + t];
    outC[t] = s / (float)B_SZ;
    if (t == 0) outCini[0] = (cini[(B_SZ - 1) * 2 + 0] + cini[(B_SZ - 1) * 2 + 1]) / (float)B_SZ;
}

// grid(64,B), block(256): per-block min/max of dists columns
__global__ void minmax_part(const float* __restrict__ dists, float* mmP) {
    int b = blockIdx.y, tid = threadIdx.x, lane = tid & 31, w = tid >> 5;
    int base = blockIdx.x * 1024;
    float mn0 = 3.4e38f, mx0 = -3.4e38f, mn1 = 3.4e38f, mx1 = -3.4e38f;
    #pragma unroll
    for (int k = 0; k < 4; ++k) {
        int i = base + k * 256 + tid;
        float2 d = *reinterpret_cast<const float2*>(dists + ((size_t)b * NPTS + i) * 2);
        mn0 = fminf(mn0, d.x); mx0 = fmaxf(mx0, d.x);
        mn1 = fminf(mn1, d.y); mx1 = fmaxf(mx1, d.y);
    }
    #pragma unroll
    for (int o = 16; o > 0; o >>= 1) {
        mn0 = fminf(mn0, __shfl_xor(mn0, o, 32)); mx0 = fmaxf(mx0, __shfl_xor(mx0, o, 32));
        mn1 = fminf(mn1, __shfl_xor(mn1, o, 32)); mx1 = fmaxf(mx1, __shfl_xor(mx1, o, 32));
    }
    __shared__ float l[8][4];
    if (lane == 0) { l[w][0] = mn0; l[w][1] = mx0; l[w][2] = mn1; l[w][3] = mx1; }
    __syncthreads();
    if (tid == 0) {
        float a = l[0][0], bm = l[0][1], c = l[0][2], d = l[0][3];
        for (int ww = 1; ww < 8; ++ww) {
            a = fminf(a, l[ww][0]); bm = fmaxf(bm, l[ww][1]);
            c = fminf(c, l[ww][2]); d = fmaxf(d, l[ww][3]);
        }
        float* o = mmP + ((size_t)b * 64 + blockIdx.x) * 4;
        o[0] = a; o[1] = bm; o[2] = c; o[3] = d;
    }
}

__global__ void minmax_fin(const float* __restrict__ mmP, float* mm) {
    int b = blockIdx.x, t = threadIdx.x;   // 64 threads
    __shared__ float l[64][4];
    const float* p = mmP + ((size_t)b * 64 + t) * 4;
    l[t][0] = p[0]; l[t][1] = p[1]; l[t][2] = p[2]; l[t][3] = p[3];
    __syncthreads();
    if (t == 0) {
        float a = l[0][0], bm = l[0][1], c = l[0][2], d = l[0][3];
        for (int k = 1; k < 64; ++k) {
            a = fminf(a, l[k][0]); bm = fmaxf(bm, l[k][1]);
            c = fminf(c, l[k][2]); d = fmaxf(d, l[k][3]);
        }
        mm[b * 4 + 0] = a; mm[b * 4 + 1] = bm; mm[b * 4 + 2] = c; mm[b * 4 + 3] = d;
    }
}

// grid(256,B), block(256): onehot + Weight from final dists/labels
__global__ void finalize_kernel(const float* __restrict__ dists, const float* __restrict__ labels,
                                const float* __restrict__ mm, float* oneh, float* wgt) {
    int b = blockIdx.y;
    int i = blockIdx.x * 256 + threadIdx.x;
    size_t gi = (size_t)b * NPTS + i;
    float2 d = *reinterpret_cast<const float2*>(dists + gi * 2);
    float lf = labels[gi];
    float mn0 = mm[b * 4 + 0], mx0 = mm[b * 4 + 1];
    float mn1 = mm[b * 4 + 2], mx1 = mm[b * 4 + 3];
    float2 oh; oh.x = 1.f - lf; oh.y = lf;
    *reinterpret_cast<float2*>(oneh + gi * 2) = oh;
    float2 wv;
    wv.x = 1.f - (d.x - mn0) / (mx0 - mn0 + 1e-7f);
    wv.y = 1.f - (d.y - mn1) / (mx1 - mn1 + 1e-7f);
    *reinterpret_cast<float2*>(wgt + gi * 2) = wv;
}

// ---------------------------------------------------------------------------
extern "C" void kernel_launch(void* const* d_in, const int* in_sizes, int n_in,
                              void* d_out, int out_size, void* d_ws, size_t ws_size,
                              hipStream_t stream) {
    (void)in_sizes; (void)n_in; (void)out_size; (void)ws_size;
    const float* F = (const float*)d_in[0];       // [8,128,256,256]
    const float* cinit = (const float*)d_in[1];   // [2,128]
    const int* n1 = (const int*)d_in[2];
    const int* n2 = (const int*)d_in[3];

    float* out = (float*)d_out;
    float* outC    = out;                 // 256
    float* outLab  = out + 256;           // 524288
    float* outOH   = out + 524544;        // 1048576
    float* outW    = out + 1573120;       // 1048576
    float* outD    = out + 2621696;       // 1048576
    float* outLP   = out + 3670272;       // 524288
    float* outCini = out + 4194560;       // 1

    float* ws      = (float*)d_ws;
    float* centers = ws;                  // 2048
    float* cn      = ws + 2048;           // 2048
    float* citer   = ws + 4096;           // 2048
    float* thr     = ws + 6144;           // 16
    float* cini    = ws + 6160;           // 16
    float* mm      = ws + 6176;           // 32
    float* p1      = ws + 6208;           // 8*512*3 = 12288
    float* p3      = ws + 18496;          // 8*32*258 = 66048
    float* mmP     = ws + 84544;          // 8*64*4 = 2048

    for (int it = 0; it < 6; ++it) {
        prep_centers<<<B_SZ, 128, 0, stream>>>(cinit, centers, cn, citer, it == 0);
        dist_kernel<<<dim3(512, B_SZ), 256, 0, stream>>>(
            F, cn, outD, outLab, (it == 1) ? outLP : (float*)nullptr, p1);
        thresh_kernel<<<B_SZ, 256, 0, stream>>>(p1, thr, n1, n2);
        flagsum_kernel<<<dim3(32, B_SZ), 256, 0, stream>>>(F, outD, thr, p3);
        citer_kernel<<<B_SZ, 256, 0, stream>>>(p3, citer, cinit, cini, it == 0);
    }
    finish_kernel<<<1, 256, 0, stream>>>(citer, cini, outC, outCini);
    minmax_part<<<dim3(64, B_SZ), 256, 0, stream>>>(outD, mmP);
    minmax_fin<<<B_SZ, 64, 0, stream>>>(mmP, mm);
    finalize_kernel<<<dim3(256, B_SZ), 256, 0, stream>>>(outD, outLab, mm, outOH, outW);
}